// EmbeddingBagCollection_84894323573299
// MI455X (gfx1250) — compile-verified
//
#include <hip/hip_runtime.h>

// EmbeddingBagCollection forward for MI455X (gfx1250).
//
// Pure-bandwidth op (0.24 FLOP/byte) -> no WMMA. CDNA5 Tensor Data Mover in
// GATHER MODE: one tensor_load_to_lds gathers up to 8 embedding rows (32-bit
// indices) = 4KB into LDS per instruction, tracked with TENSORcnt. Each wave
// owns one (feature, bag), double-buffers TDM gathers in LDS, and accumulates
// one float4 per lane (lane l = dims 4l..4l+3).

#define Ff 8
#define Bb 4096
#define Vv 100000
#define Dd 128
#define Nn 81920

#define WAVES_PER_BLOCK 4
#define CHUNK 8                        // rows per TDM gather (32-bit idx mode max = 8)
#define BUF_F4 (CHUNK * 32)            // 256 float4 = 4 KB per pipeline buffer
#define BUF_BYTES (BUF_F4 * 16)

typedef __attribute__((ext_vector_type(4))) unsigned int v4u;
typedef __attribute__((ext_vector_type(8))) unsigned int v8u;

// Issue one TDM gather of `cnt` (1..8) rows of the table at tab_addr
// (row k = vals[clamp(base+k)]) into the LDS buffer at byte offset lds_off.
// `bufptr` is the same buffer as a pointer: it is passed into the asm as an
// (unreferenced) operand purely so the buffer's address ESCAPES -- without it
// LLVM proves the LDS array is never written and deletes all reads of it.
// D# layout per CDNA5 ISA (gather mode).
__device__ __forceinline__ void tdm_gather_issue(unsigned lds_off,
                                                 float4* bufptr,
                                                 unsigned long long tab_addr,
                                                 const int* __restrict__ vals,
                                                 int base, int cnt, int len) {
    // Clamped offsets: always in-bounds (len >= 1 whenever we issue); TDM
    // ignores indices past tile_dim1 = cnt, so tail duplicates are harmless.
    const int last = len - 1;
    int o0 = base + 0; o0 = (o0 > last) ? last : o0;
    int o1 = base + 1; o1 = (o1 > last) ? last : o1;
    int o2 = base + 2; o2 = (o2 > last) ? last : o2;
    int o3 = base + 3; o3 = (o3 > last) ? last : o3;
    int o4 = base + 4; o4 = (o4 > last) ? last : o4;
    int o5 = base + 5; o5 = (o5 > last) ? last : o5;
    int o6 = base + 6; o6 = (o6 > last) ? last : o6;
    int o7 = base + 7; o7 = (o7 > last) ? last : o7;

    // Group 0: count=1 | gather_index_size=1 (32-bit) | gather_mode=1,
    //          lds_addr, global_addr[56:0], type=2.
    v4u g0;
    g0.x = 1u | (1u << 30) | (1u << 31);
    g0.y = lds_off;                                      // static LDS at offset 0
    g0.z = (unsigned)(tab_addr & 0xffffffffu);
    g0.w = (unsigned)((tab_addr >> 32) & 0x01ffffffu) | (2u << 30);

    // Group 1: workgroup_mask=0, data_size=2 (4B), no barrier/iterate/pad.
    //   tensor_dim0=128 [79:48], tensor_dim1=V [111:80], tile_dim0=128 [127:112],
    //   tile_dim1=cnt (#valid indices) [143:128], tensor_dim0_stride=128 [207:160].
    v8u g1;
    g1.s0 = (2u << 16);
    g1.s1 = ((unsigned)Dd << 16);
    g1.s2 = (((unsigned)Vv & 0xffffu) << 16);
    g1.s3 = ((unsigned)Vv >> 16) | ((unsigned)Dd << 16);
    g1.s4 = (unsigned)cnt;
    g1.s5 = (unsigned)Dd;
    g1.s6 = 0u;
    g1.s7 = 0u;

    // Groups 2/3: eight 32-bit row indices.
    v4u g2; v4u g3;
    g2.x = (unsigned)vals[o0]; g2.y = (unsigned)vals[o1];
    g2.z = (unsigned)vals[o2]; g2.w = (unsigned)vals[o3];
    g3.x = (unsigned)vals[o4]; g3.y = (unsigned)vals[o5];
    g3.z = (unsigned)vals[o6]; g3.w = (unsigned)vals[o7];

    // %4 (bufptr) is intentionally unreferenced: address escape only.
    asm volatile("tensor_load_to_lds %0, %1, %2, %3"
                 :: "s"(g0), "s"(g1), "s"(g2), "s"(g3), "v"(bufptr)
                 : "memory");
}

__global__ __launch_bounds__(WAVES_PER_BLOCK * 32)
void ebc_fwd_tdm(const int* __restrict__ values,    // [F, N]
                 const int* __restrict__ offsets,   // [F, B+1]
                 const float* __restrict__ tables,  // [F, V, D]
                 float* __restrict__ out) {         // [B, F*D]
    // Typed shared array so addrspace(3) survives -> real ds_load_b128.
    __shared__ float4 smem4[WAVES_PER_BLOCK * 2 * BUF_F4];   // 32 KB

    const int lane = (int)(threadIdx.x & 31u);
    const int wid  = __builtin_amdgcn_readfirstlane((int)(threadIdx.x >> 5));
    const int gw   = (int)blockIdx.x * WAVES_PER_BLOCK + wid;  // [0, F*B)
    const int f    = gw >> 12;          // / 4096
    const int b    = gw & (Bb - 1);

    const int start = offsets[f * (Bb + 1) + b];
    const int end   = offsets[f * (Bb + 1) + b + 1];
    const int len   = end - start;

    const int* vals = values + (size_t)f * Nn + start;
    const unsigned long long tab =
        (unsigned long long)(const void*)(tables + (size_t)f * Vv * Dd);

    const unsigned lds_wave = (unsigned)(wid * 2 * BUF_BYTES);
    const int      f4_wave  = wid * 2 * BUF_F4;
    float4* buf0 = &smem4[f4_wave];
    float4* buf1 = &smem4[f4_wave + BUF_F4];

    float ax = 0.f, ay = 0.f, az = 0.f, aw = 0.f;

    const int nch = (len + CHUNK - 1) / CHUNK;

    // Prologue: fill both pipeline buffers.
    if (nch > 0) {
        int c0 = (len < CHUNK) ? len : CHUNK;
        tdm_gather_issue(lds_wave, buf0, tab, vals, 0, c0, len);
    }
    if (nch > 1) {
        int rem = len - CHUNK;
        int c1 = (rem < CHUNK) ? rem : CHUNK;
        tdm_gather_issue(lds_wave + BUF_BYTES, buf1, tab, vals, CHUNK, c1, len);
    }

    for (int c = 0; c < nch; ++c) {
        // Wait for the buffer holding chunk c (keep 1 gather in flight if any).
        if (c + 1 < nch) {
            asm volatile("s_wait_tensorcnt 1" ::: "memory");
        } else {
            asm volatile("s_wait_tensorcnt 0" ::: "memory");
        }

        const int rem = len - c * CHUNK;
        const int cnt = (rem < CHUNK) ? rem : CHUNK;
        const int bufbase = f4_wave + (c & 1) * BUF_F4 + lane;  // row r at +32*r

        if (cnt == CHUNK) {
            // Common case: 8 back-to-back ds_load_b128, then accumulate.
            float4 v0 = smem4[bufbase + 0 * 32];
            float4 v1 = smem4[bufbase + 1 * 32];
            float4 v2 = smem4[bufbase + 2 * 32];
            float4 v3 = smem4[bufbase + 3 * 32];
            float4 v4 = smem4[bufbase + 4 * 32];
            float4 v5 = smem4[bufbase + 5 * 32];
            float4 v6 = smem4[bufbase + 6 * 32];
            float4 v7 = smem4[bufbase + 7 * 32];
            ax += v0.x + v1.x + v2.x + v3.x + v4.x + v5.x + v6.x + v7.x;
            ay += v0.y + v1.y + v2.y + v3.y + v4.y + v5.y + v6.y + v7.y;
            az += v0.z + v1.z + v2.z + v3.z + v4.z + v5.z + v6.z + v7.z;
            aw += v0.w + v1.w + v2.w + v3.w + v4.w + v5.w + v6.w + v7.w;
        } else {
            for (int r = 0; r < cnt; ++r) {
                float4 v = smem4[bufbase + r * 32];     // ds_load_b128
                ax += v.x; ay += v.y; az += v.z; aw += v.w;
            }
        }

        // LDS reads must land in VGPRs before TDM overwrites this buffer.
        asm volatile("s_wait_dscnt 0" ::: "memory");

        const int nxt = c + 2;
        if (nxt < nch) {
            int r2 = len - nxt * CHUNK;
            int c2 = (r2 < CHUNK) ? r2 : CHUNK;
            tdm_gather_issue(lds_wave + (unsigned)((c & 1) * BUF_BYTES),
                             (c & 1) ? buf1 : buf0,
                             tab, vals, nxt * CHUNK, c2, len);
        }
    }

    // One fully-coalesced 512B store per wave: out[b, f*128 + 4*lane .. +3].
    float4 res;
    res.x = ax; res.y = ay; res.z = az; res.w = aw;
    float* op = out + (size_t)b * (Ff * Dd) + (size_t)f * Dd + lane * 4;
    *(float4*)op = res;
}

extern "C" void kernel_launch(void* const* d_in, const int* in_sizes, int n_in,
                              void* d_out, int out_size, void* d_ws, size_t ws_size,
                              hipStream_t stream) {
    const int*   values  = (const int*)d_in[0];   // [F*N]
    const int*   offsets = (const int*)d_in[1];   // [F*(B+1)]
    const float* tables  = (const float*)d_in[2]; // [F*V*D]
    float*       out     = (float*)d_out;         // [B*F*D]

    dim3 grid((Ff * Bb) / WAVES_PER_BLOCK);
    dim3 block(WAVES_PER_BLOCK * 32);
    hipLaunchKernelGGL(ebc_fwd_tdm, grid, block, 0, stream,
                       values, offsets, tables, out);
    (void)in_sizes; (void)n_in; (void)out_size; (void)d_ws; (void)ws_size;
}